// UncertaintyEvalWrapper_53266184405033
// MI455X (gfx1250) — compile-verified
//
#include <hip/hip_runtime.h>
#include <hip/hip_bf16.h>

// ---------------------------------------------------------------------------
// RANSAC voting (PVNet) for MI455X / gfx1250.
// Voting inner loop mapped onto V_WMMA_F32_16X16X4_F32 (two K=4 GEMMs per
// 16x16 hypothesis x pixel tile). wave32, one wave per (b, v, hyp-tile).
// ---------------------------------------------------------------------------

#define B_    2
#define H_    64
#define W_    64
#define P_    (H_ * W_)      // 4096 pixels
#define VN_   9
#define HN_R  512            // hypotheses per RANSAC round
#define HN_D  4096           // hypotheses for distribution estimation
#define NIT_  21             // MAX_ITER + 1 scan steps
#define T2_   (0.99f * 0.99f)

typedef __attribute__((ext_vector_type(2))) float v2f;
typedef __attribute__((ext_vector_type(8))) float v8f;

// ---- workspace layout (bytes), all regions 256B separated -----------------
constexpr size_t OFF_FGMASK = 0;                                   // B*P int
constexpr size_t OFF_FGLIST = OFF_FGMASK + (size_t)B_ * P_ * 4;    // B*P int
constexpr size_t OFF_FGCNT  = OFF_FGLIST + (size_t)B_ * P_ * 4;    // B int
constexpr size_t OFF_ACTIVE = OFF_FGCNT + 256;                     // B int
constexpr size_t OFF_HYPNUM = OFF_ACTIVE + 256;                    // B float
constexpr size_t OFF_BRATIO = OFF_HYPNUM + 256;                    // B*VN float
constexpr size_t OFF_BPTS   = OFF_BRATIO + 256;                    // B*VN*2 float
constexpr size_t OFF_B1     = OFF_BPTS + 256;                      // B*VN*P float4
constexpr size_t OFF_B2     = OFF_B1 + (size_t)B_ * VN_ * P_ * 16; // B*VN*P float4
constexpr size_t OFF_HYPS   = OFF_B2 + (size_t)B_ * VN_ * P_ * 16; // B*HN_D*VN*2 f32
constexpr size_t OFF_CNTS   = OFF_HYPS + (size_t)B_ * HN_D * VN_ * 2 * 4; // B*HN_D*VN int
// total ~3.3 MB

// ---- deterministic hash RNG ----------------------------------------------
__device__ __forceinline__ unsigned int hash_u32(unsigned int x) {
  x ^= x >> 16; x *= 0x7feb352dU;
  x ^= x >> 15; x *= 0x846ca68bU;
  x ^= x >> 16;
  return x;
}
__device__ __forceinline__ unsigned int rng4(unsigned int seed, unsigned int a,
                                             unsigned int b, unsigned int c) {
  return hash_u32(seed ^ hash_u32(a + 0x9e3779b9u * hash_u32(b + 0x85ebca6bu * hash_u32(c + 1u))));
}

// ---------------------------------------------------------------------------
__global__ void k_init(char* ws) {
  int t = threadIdx.x;
  int*   active = (int*)(ws + OFF_ACTIVE);
  float* hypnum = (float*)(ws + OFF_HYPNUM);
  float* bratio = (float*)(ws + OFF_BRATIO);
  float* bpts   = (float*)(ws + OFF_BPTS);
  if (t < B_)          { active[t] = 1; hypnum[t] = 0.f; }
  if (t < B_ * VN_)      bratio[t] = 0.f;
  if (t < B_ * VN_ * 2)  bpts[t]   = 0.f;
}

__global__ void k_fgmask(const float* __restrict__ seg, char* ws) {
  int g = blockIdx.x * blockDim.x + threadIdx.x;
  if (g >= B_ * P_) return;
  int b = g / P_, p = g % P_;
  int fg = seg[b * 2 * P_ + P_ + p] > seg[b * 2 * P_ + p];
  ((int*)(ws + OFF_FGMASK))[g] = fg;
}

__global__ void k_compact(char* ws) {  // deterministic serial compaction per batch
  int b = blockIdx.x;
  const int* m = (const int*)(ws + OFF_FGMASK) + b * P_;
  int* lst = (int*)(ws + OFF_FGLIST) + b * P_;
  int c = 0;
  for (int p = 0; p < P_; ++p)
    if (m[p]) lst[c++] = p;
  ((int*)(ws + OFF_FGCNT))[b] = c;
}

// Build the two K=4 "B-matrix" tables. fg baked in: bg column forces dot = -1.
__global__ void k_prepB(const float* __restrict__ vert, char* ws) {
  int g = blockIdx.x * blockDim.x + threadIdx.x;
  if (g >= B_ * VN_ * P_) return;
  int b = g / (VN_ * P_);
  int r = g % (VN_ * P_);
  int v = r / P_, p = r % P_;
  int fg = ((const int*)(ws + OFF_FGMASK))[b * P_ + p];
  float x = (float)(p & (W_ - 1));
  float y = (float)(p >> 6);
  float dx = vert[(size_t)b * 2 * VN_ * P_ + (2 * v + 0) * P_ + p];
  float dy = vert[(size_t)b * 2 * VN_ * P_ + (2 * v + 1) * P_ + p];
  float4 b1, b2;
  if (fg) {
    float s = T2_ * (dx * dx + dy * dy);
    b1 = make_float4(dx, dy, -(x * dx + y * dy), 0.f);
    b2 = make_float4(-2.f * x * s, -2.f * y * s, s, (x * x + y * y) * s);
  } else {
    b1 = make_float4(0.f, 0.f, -1.f, 0.f);
    b2 = make_float4(0.f, 0.f, 0.f, 0.f);
  }
  ((float4*)(ws + OFF_B1))[(size_t)(b * VN_ + v) * P_ + p] = b1;
  ((float4*)(ws + OFF_B2))[(size_t)(b * VN_ + v) * P_ + p] = b2;
}

// Sample foreground pixel pairs, compute line intersections.
__global__ void k_gen(const float* __restrict__ vert, char* ws,
                      unsigned int seed, int hn, int checkActive) {
  int g = blockIdx.x * blockDim.x + threadIdx.x;
  if (g >= B_ * hn * VN_) return;
  int b = g / (hn * VN_);
  int r = g % (hn * VN_);
  int h = r / VN_, v = r % VN_;
  if (checkActive && !((const int*)(ws + OFF_ACTIVE))[b]) return;
  int tn = ((const int*)(ws + OFF_FGCNT))[b];
  if (tn <= 0) tn = 1;
  const int* lst = (const int*)(ws + OFF_FGLIST) + b * P_;
  int p0 = lst[rng4(seed, (unsigned)(b * 131072 + h), (unsigned)v, 0u) % (unsigned)tn] & (P_ - 1);
  int p1 = lst[rng4(seed, (unsigned)(b * 131072 + h), (unsigned)v, 1u) % (unsigned)tn] & (P_ - 1);
  float p0x = (float)(p0 & 63), p0y = (float)(p0 >> 6);
  float p1x = (float)(p1 & 63), p1y = (float)(p1 >> 6);
  size_t vb = (size_t)b * 2 * VN_ * P_;
  float d0x = vert[vb + (2 * v + 0) * P_ + p0];
  float d0y = vert[vb + (2 * v + 1) * P_ + p0];
  float d1x = vert[vb + (2 * v + 0) * P_ + p1];
  float d1y = vert[vb + (2 * v + 1) * P_ + p1];
  float denom = d0x * d1y - d0y * d1x;
  float numv  = (p1x - p0x) * d1y - (p1y - p0y) * d1x;
  float t = (fabsf(denom) > 1e-6f) ? (numv / denom) : 0.f;
  float* hyp = (float*)(ws + OFF_HYPS) + ((size_t)(b * HN_D + h) * VN_ + v) * 2;
  hyp[0] = p0x + t * d0x;
  hyp[1] = p0y + t * d0y;
}

// WMMA voting: one wave per (b, v, 16-hypothesis tile); streams 256 pixel
// tiles, two V_WMMA_F32_16X16X4_F32 per tile (dot and scaled-distance forms).
__global__ __launch_bounds__(32) void k_vote(char* ws, int hn, int checkActive) {
  int nht = hn >> 4;
  int wid = blockIdx.x;
  int b  = wid / (VN_ * nht);
  int rr = wid % (VN_ * nht);
  int v  = rr / nht;
  int ht = rr % nht;
  if (checkActive && !((const int*)(ws + OFF_ACTIVE))[b]) return;

  int lane = threadIdx.x;
  int row  = lane & 15;
  int h    = ht * 16 + row;
  const float* hyp = (const float*)(ws + OFF_HYPS) + ((size_t)(b * HN_D + h) * VN_ + v) * 2;
  float hx = hyp[0], hy = hyp[1];

  // A tiles (16x4 f32): VGPR0 = K0 (lanes 0-15) / K2 (lanes 16-31), VGPR1 = K1/K3
  v2f a_dot, a_q;
  if (lane < 16) { a_dot = (v2f){hx, hy};   a_q = (v2f){hx, hy}; }
  else           { a_dot = (v2f){1.f, 0.f}; a_q = (v2f){hx * hx + hy * hy, 1.f}; }

  const float4* pB1 = (const float4*)(ws + OFF_B1) + (size_t)(b * VN_ + v) * P_;
  const float4* pB2 = (const float4*)(ws + OFF_B2) + (size_t)(b * VN_ + v) * P_;
  int col = lane & 15;

  int cnt[8];
#pragma unroll
  for (int i = 0; i < 8; ++i) cnt[i] = 0;

  for (int pt = 0; pt < P_ / 16; ++pt) {
    __builtin_prefetch(&pB1[(pt + 8) * 16 + col], 0, 1);  // global_prefetch_b8
    float4 c1 = pB1[pt * 16 + col];
    float4 c2 = pB2[pt * 16 + col];
    // B tiles (4x16 f32): VGPR0 = K0 row (lanes 0-15) / K2 row (lanes 16-31)
    v2f b_dot = (lane < 16) ? (v2f){c1.x, c1.y} : (v2f){c1.z, c1.w};
    v2f b_q   = (lane < 16) ? (v2f){c2.x, c2.y} : (v2f){c2.z, c2.w};
    v8f cd = {}; v8f cq = {};
    cd = __builtin_amdgcn_wmma_f32_16x16x4_f32(false, a_dot, false, b_dot,
                                               (short)0, cd, false, false);
    cq = __builtin_amdgcn_wmma_f32_16x16x4_f32(false, a_q, false, b_q,
                                               (short)0, cq, false, false);
#pragma unroll
    for (int i = 0; i < 8; ++i) {
      float d = cd[i];
      cnt[i] += (d > 0.f && d * d > cq[i]) ? 1 : 0;
    }
  }

  // Reduce over the 16 lanes holding each row (C layout: VGPR i = row i / i+8)
#pragma unroll
  for (int i = 0; i < 8; ++i) {
    cnt[i] += __shfl_xor(cnt[i], 1, 32);
    cnt[i] += __shfl_xor(cnt[i], 2, 32);
    cnt[i] += __shfl_xor(cnt[i], 4, 32);
    cnt[i] += __shfl_xor(cnt[i], 8, 32);
  }
  if ((lane & 15) == 0) {
    int half = lane >> 4;  // 0 -> rows 0..7, 1 -> rows 8..15
    int* cnts = (int*)(ws + OFF_CNTS) + (size_t)b * HN_D * VN_;
#pragma unroll
    for (int i = 0; i < 8; ++i)
      cnts[(ht * 16 + half * 8 + i) * VN_ + v] = cnt[i];
  }
}

// argmax over counts per (b,v); update best if improved and active
__global__ void k_best(char* ws, int hn) {
  int bv = blockIdx.x;
  int b = bv / VN_, v = bv % VN_;
  if (!((const int*)(ws + OFF_ACTIVE))[b]) return;
  __shared__ int sc[256], si[256];
  int tid = threadIdx.x;
  const int* cnts = (const int*)(ws + OFF_CNTS) + (size_t)b * HN_D * VN_;
  int best = -1, bidx = 0;
  for (int h = tid; h < hn; h += 256) {
    int c = cnts[h * VN_ + v];
    if (c > best || (c == best && h < bidx)) { best = c; bidx = h; }
  }
  sc[tid] = best; si[tid] = bidx;
  __syncthreads();
  for (int s = 128; s > 0; s >>= 1) {
    if (tid < s) {
      if (sc[tid + s] > sc[tid] ||
          (sc[tid + s] == sc[tid] && si[tid + s] < si[tid])) {
        sc[tid] = sc[tid + s]; si[tid] = si[tid + s];
      }
    }
    __syncthreads();
  }
  if (tid == 0) {
    int tn = ((const int*)(ws + OFF_FGCNT))[b];
    float ratio = (float)sc[0] / (float)(tn > 0 ? tn : 1);
    float* bratio = (float*)(ws + OFF_BRATIO);
    float* bpts   = (float*)(ws + OFF_BPTS);
    if (ratio > bratio[bv]) {
      bratio[bv] = ratio;
      const float* hp = (const float*)(ws + OFF_HYPS) +
                        ((size_t)(b * HN_D + si[0]) * VN_ + v) * 2;
      bpts[bv * 2 + 0] = hp[0];
      bpts[bv * 2 + 1] = hp[1];
    }
  }
}

__global__ void k_update(char* ws) {
  int b = blockIdx.x;
  int* active = (int*)(ws + OFF_ACTIVE);
  if (!active[b]) return;
  float* hypnum = (float*)(ws + OFF_HYPNUM);
  const float* bratio = (const float*)(ws + OFF_BRATIO);
  hypnum[b] += (float)HN_R;
  float minr = 1e30f;
  for (int v = 0; v < VN_; ++v) minr = fminf(minr, bratio[b * VN_ + v]);
  float conf = 1.f - powf(1.f - minr * minr, hypnum[b]);
  if (conf > 0.99f) active[b] = 0;
}

// weighted least-squares refit over inliers of winning point -> mean
__global__ void k_refit(const float* __restrict__ vert, char* ws, float* out) {
  int bv = blockIdx.x;
  int b = bv / VN_, v = bv % VN_;
  int tid = threadIdx.x;
  const float* bpts = (const float*)(ws + OFF_BPTS);
  float hx = bpts[bv * 2 + 0], hy = bpts[bv * 2 + 1];
  const int* fgm = (const int*)(ws + OFF_FGMASK) + b * P_;
  size_t vb = (size_t)b * 2 * VN_ * P_;
  float a00 = 0.f, a01 = 0.f, a11 = 0.f, s0 = 0.f, s1 = 0.f;
  for (int p = tid; p < P_; p += 256) {
    if (!fgm[p]) continue;
    float x = (float)(p & 63), y = (float)(p >> 6);
    float dx = vert[vb + (2 * v + 0) * P_ + p];
    float dy = vert[vb + (2 * v + 1) * P_ + p];
    float ex = hx - x, ey = hy - y;
    float dot = ex * dx + ey * dy;
    float d2 = ex * ex + ey * ey, dn2 = dx * dx + dy * dy;
    if (dot > 0.f && dot * dot > T2_ * d2 * dn2) {
      float nx = dy, ny = -dx;
      float bb = nx * x + ny * y;
      a00 += nx * nx; a01 += nx * ny; a11 += ny * ny;
      s0 += nx * bb; s1 += ny * bb;
    }
  }
  __shared__ float sm[5][256];
  sm[0][tid] = a00; sm[1][tid] = a01; sm[2][tid] = a11;
  sm[3][tid] = s0;  sm[4][tid] = s1;
  __syncthreads();
  for (int s = 128; s > 0; s >>= 1) {
    if (tid < s)
      for (int k = 0; k < 5; ++k) sm[k][tid] += sm[k][tid + s];
    __syncthreads();
  }
  if (tid == 0) {
    float A00 = sm[0][0] + 1e-8f, A01 = sm[1][0], A11 = sm[2][0] + 1e-8f;
    float B0 = sm[3][0], B1v = sm[4][0];
    float det = A00 * A11 - A01 * A01;
    out[bv * 2 + 0] = (A11 * B0 - A01 * B1v) / det;
    out[bv * 2 + 1] = (A00 * B1v - A01 * B0) / det;
  }
}

// weighted covariance of distribution hypotheses about the mean -> var
__global__ void k_cov(char* ws, float* out) {
  int bv = blockIdx.x;
  int b = bv / VN_, v = bv % VN_;
  int tid = threadIdx.x;
  float tn = (float)max(((const int*)(ws + OFF_FGCNT))[b], 1);
  const int* cnts = (const int*)(ws + OFF_CNTS) + (size_t)b * HN_D * VN_;
  const float* hyps = (const float*)(ws + OFF_HYPS);
  float mx = out[bv * 2 + 0], my = out[bv * 2 + 1];
  __shared__ float sm[4][256];
  // phase 1: max ratio
  float mr = -1e30f;
  for (int i = tid; i < HN_D; i += 256)
    mr = fmaxf(mr, (float)cnts[i * VN_ + v] / tn);
  sm[0][tid] = mr;
  __syncthreads();
  for (int s = 128; s > 0; s >>= 1) {
    if (tid < s) sm[0][tid] = fmaxf(sm[0][tid], sm[0][tid + s]);
    __syncthreads();
  }
  float thresh = sm[0][0] - 0.1f;
  __syncthreads();
  // phase 2: weighted covariance
  float c00 = 0.f, c01 = 0.f, c11 = 0.f, wsum = 0.f;
  for (int i = tid; i < HN_D; i += 256) {
    float r = (float)cnts[i * VN_ + v] / tn;
    r = (r < thresh) ? 0.f : r;
    if (r > 0.f) {
      const float* hp = hyps + ((size_t)(b * HN_D + i) * VN_ + v) * 2;
      float d0 = hp[0] - mx, d1 = hp[1] - my;
      c00 += r * d0 * d0; c01 += r * d0 * d1; c11 += r * d1 * d1;
      wsum += r;
    }
  }
  sm[0][tid] = c00; sm[1][tid] = c01; sm[2][tid] = c11; sm[3][tid] = wsum;
  __syncthreads();
  for (int s = 128; s > 0; s >>= 1) {
    if (tid < s)
      for (int k = 0; k < 4; ++k) sm[k][tid] += sm[k][tid + s];
    __syncthreads();
  }
  if (tid == 0) {
    float dn = sm[3][0] + 1e-3f;
    float* var = out + B_ * VN_ * 2 + bv * 4;
    var[0] = sm[0][0] / dn; var[1] = sm[1][0] / dn;
    var[2] = sm[1][0] / dn; var[3] = sm[2][0] / dn;
  }
}

// ---------------------------------------------------------------------------
extern "C" void kernel_launch(void* const* d_in, const int* in_sizes, int n_in,
                              void* d_out, int out_size, void* d_ws, size_t ws_size,
                              hipStream_t stream) {
  const float* seg  = (const float*)d_in[0];   // [B,2,H,W]
  const float* vert = (const float*)d_in[1];   // [B,2*VN,H,W]
  float* out = (float*)d_out;                  // mean (36) ++ var (72)
  char* ws = (char*)d_ws;                      // needs ~3.3 MB

  k_init<<<1, 64, 0, stream>>>(ws);
  k_fgmask<<<(B_ * P_ + 255) / 256, 256, 0, stream>>>(seg, ws);
  k_compact<<<B_, 1, 0, stream>>>(ws);
  k_prepB<<<(B_ * VN_ * P_ + 255) / 256, 256, 0, stream>>>(vert, ws);

  for (int r = 0; r < NIT_; ++r) {
    unsigned int seed = 0x9E3779B9u * (unsigned)(r + 1) ^ 0xA511E9B3u;
    k_gen<<<(B_ * HN_R * VN_ + 255) / 256, 256, 0, stream>>>(vert, ws, seed, HN_R, 1);
    k_vote<<<B_ * VN_ * (HN_R / 16), 32, 0, stream>>>(ws, HN_R, 1);
    k_best<<<B_ * VN_, 256, 0, stream>>>(ws, HN_R);
    k_update<<<B_, 1, 0, stream>>>(ws);
  }

  k_refit<<<B_ * VN_, 256, 0, stream>>>(vert, ws, out);

  unsigned int dseed = 0x85EBCA6Bu;
  k_gen<<<(B_ * HN_D * VN_ + 255) / 256, 256, 0, stream>>>(vert, ws, dseed, HN_D, 0);
  k_vote<<<B_ * VN_ * (HN_D / 16), 32, 0, stream>>>(ws, HN_D, 0);
  k_cov<<<B_ * VN_, 256, 0, stream>>>(ws, out);
}